// Transpooling_44985487458919
// MI455X (gfx1250) — compile-verified
//
#include <hip/hip_runtime.h>

typedef __attribute__((ext_vector_type(2))) float v2f;
typedef __attribute__((ext_vector_type(8))) float v8f;

#define NN 512
#define MM 31
#define T  32        // tokens = M+1
#define E  64        // INP == EMB == ATT
#define SELF_LOOP 200

// Low 32 bits of a generic pointer to LDS == LDS byte offset (ISA 10.2:
// the LDS aperture uses the low 32 address bits as the LDS address).
__device__ __forceinline__ unsigned lds_off(const void* p) {
    return (unsigned)(size_t)p;
}

__global__ __launch_bounds__(128)
void transpool_kernel(const float* __restrict__ h,
                      const float* __restrict__ msg,
                      const int*   __restrict__ msg_type,
                      const int*   __restrict__ r_label_node,
                      const int*   __restrict__ r_label_msg,
                      const float* __restrict__ W_self,
                      const float* __restrict__ Qw,
                      const float* __restrict__ Kw,
                      const float* __restrict__ Vw,
                      const float* __restrict__ ffn_w,
                      const float* __restrict__ ffn_b,
                      float* __restrict__ out)
{
    __shared__ __align__(16) float sx[T][E];   // tokens (row 0 = curr)
    __shared__ __align__(16) float sh[E];      // staged h row
    __shared__ float sq[T][E];
    __shared__ float sk[T][E];
    __shared__ float sv[T][E];
    __shared__ float sS[T][T];                 // scores (pre-softmax)
    __shared__ float srow[T];                  // att[n,0,:] after softmax over q
    __shared__ float spooled[E];
    __shared__ int   qvIdx[T];
    __shared__ int   kIdx[T];

    const int n   = blockIdx.x;
    const int tid = threadIdx.x;

    // ---- Phase A: async global->LDS staging (CDNA5 ASYNCcnt path) ----
    // h row: 64 floats = 16 x B128 chunks
    if (tid < E / 4) {
        const float* gsrc = h + (size_t)n * E + tid * 4;
        unsigned dst = lds_off(&sh[tid * 4]);
        asm volatile("global_load_async_to_lds_b128 %0, %1, off"
                     :: "v"(dst), "v"(gsrc) : "memory");
    }
    // msg rows: 31*64 floats = 496 x B128 chunks, no VGPR round-trip
    for (int p = tid; p < MM * (E / 4); p += 128) {
        int m = p >> 4, c = p & 15;
        const float* gsrc = msg + ((size_t)n * MM + m) * E + c * 4;
        unsigned dst = lds_off(&sx[m + 1][c * 4]);
        asm volatile("global_load_async_to_lds_b128 %0, %1, off"
                     :: "v"(dst), "v"(gsrc) : "memory");
    }
    if (tid >= 64 && tid < 64 + T) {
        int m = tid - 64;
        if (m == 0) { qvIdx[0] = r_label_node[n]; kIdx[0] = SELF_LOOP; }
        else {
            qvIdx[m] = r_label_msg[(size_t)n * MM + m - 1];
            kIdx[m]  = msg_type   [(size_t)n * MM + m - 1];
        }
    }
    asm volatile("s_wait_asynccnt 0x0" ::: "memory");
    __syncthreads();

    // ---- Phase B: curr = h[n] @ W_self -> sx[0][:]; prefetch gathered weights ----
    if (tid < T) {   // warm L2/L0 for the relation-indexed gather below
        __builtin_prefetch(Qw + (size_t)qvIdx[tid] * E * E, 0, 0);
        __builtin_prefetch(Kw + (size_t)kIdx[tid]  * E * E, 0, 0);
        __builtin_prefetch(Vw + (size_t)qvIdx[tid] * E * E, 0, 0);
    }
    if (tid < E) {
        float acc = 0.f;
        #pragma unroll 8
        for (int e = 0; e < E; ++e) acc = fmaf(sh[e], W_self[e * E + tid], acc);
        sx[0][tid] = acc;
    }
    __syncthreads();

    // ---- Phase C: relation-indexed projections (irregular gather -> VALU) ----
    for (int p = tid; p < T * E; p += 128) {
        int m = p >> 6, a = p & 63;
        const float* Qp = Qw + (size_t)qvIdx[m] * E * E + a;
        const float* Kp = Kw + (size_t)kIdx[m]  * E * E + a;
        const float* Vp = Vw + (size_t)qvIdx[m] * E * E + a;
        float aq = 0.f, ak = 0.f, av = 0.f;
        #pragma unroll 4
        for (int e = 0; e < E; ++e) {
            float xe = sx[m][e];
            aq = fmaf(xe, Qp[e * E], aq);
            ak = fmaf(xe, Kp[e * E], ak);
            av = fmaf(xe, Vp[e * E], av);
        }
        sq[m][a] = aq; sk[m][a] = ak; sv[m][a] = av;
    }
    __syncthreads();

    // ---- Phase D: scores = q @ k^T / sqrt(64) via V_WMMA_F32_16X16X4_F32 ----
    // 4 waves -> 4 output tiles of 16x16; K=64 in 16 steps of 4.
    {
        int wv   = tid >> 5;              // wave id 0..3
        int lane = tid & 31;
        int ti   = (wv >> 1) << 4;        // row-tile base
        int tj   = (wv & 1) << 4;         // col-tile base
        int half = lane >> 4;             // 0: K{0,1}, 1: K{2,3}
        int l16  = lane & 15;

        v8f c = {0.f, 0.f, 0.f, 0.f, 0.f, 0.f, 0.f, 0.f};
        #pragma unroll
        for (int kk = 0; kk < 16; ++kk) {
            int kb = kk * 4 + half * 2;
            v2f a, b;
            a.x = sq[ti + l16][kb];  a.y = sq[ti + l16][kb + 1];   // A: 16x4
            b.x = sk[tj + l16][kb];  b.y = sk[tj + l16][kb + 1];   // B: 4x16 (= k^T cols)
            c = __builtin_amdgcn_wmma_f32_16x16x4_f32(
                    false, a, false, b, (short)0, c, false, false);
        }
        #pragma unroll
        for (int r = 0; r < 8; ++r)
            sS[ti + half * 8 + r][tj + l16] = c[r] * 0.125f;       // 1/sqrt(ATT)
    }
    __syncthreads();

    // ---- Phase E: softmax over q axis (columns of S), keep only row 0 ----
    if (tid < T) {
        float mx = -3.402823466e38f;
        #pragma unroll 4
        for (int q = 0; q < T; ++q) mx = fmaxf(mx, sS[q][tid]);
        float den = 0.f;
        #pragma unroll 4
        for (int q = 0; q < T; ++q) den += __expf(sS[q][tid] - mx);
        srow[tid] = __expf(sS[0][tid] - mx) / den;
    }
    __syncthreads();

    // ---- Phase F: pooled = att_row0 @ v  (1x32 @ 32x64) ----
    if (tid < E) {
        float acc = 0.f;
        #pragma unroll 4
        for (int k = 0; k < T; ++k) acc = fmaf(srow[k], sv[k][tid], acc);
        spooled[tid] = acc;
    }
    __syncthreads();

    // ---- Phase G: out = pooled @ ffn_w + ffn_b ----
    if (tid < E) {
        float acc = ffn_b[tid];
        #pragma unroll 4
        for (int a = 0; a < E; ++a) acc = fmaf(spooled[a], ffn_w[a * E + tid], acc);
        out[(size_t)n * E + tid] = acc;
    }
}

extern "C" void kernel_launch(void* const* d_in, const int* in_sizes, int n_in,
                              void* d_out, int out_size, void* d_ws, size_t ws_size,
                              hipStream_t stream) {
    const float* h        = (const float*)d_in[0];
    const float* msg      = (const float*)d_in[1];
    const int*   msg_type = (const int*)  d_in[2];
    const int*   rln      = (const int*)  d_in[3];
    const int*   rlm      = (const int*)  d_in[4];
    const float* W_self   = (const float*)d_in[5];
    const float* Qw       = (const float*)d_in[6];
    const float* Kw       = (const float*)d_in[7];
    const float* Vw       = (const float*)d_in[8];
    const float* fw       = (const float*)d_in[9];
    const float* fb       = (const float*)d_in[10];
    float* out = (float*)d_out;

    transpool_kernel<<<NN, 128, 0, stream>>>(h, msg, msg_type, rln, rlm,
                                             W_self, Qw, Kw, Vw, fw, fb, out);
}